// EdgeConv_60610578481267
// MI455X (gfx1250) — compile-verified
//
#include <hip/hip_runtime.h>

typedef __attribute__((ext_vector_type(2))) float v2f;
typedef __attribute__((ext_vector_type(8))) float v8f;

#define C_IN 64
#define TWO_C 128
#define C_OUT 64
#define TILE 16
#define NT 4                      // 4 n-tiles of 16 output channels
#define KSTEPS 16                 // 64 features / K=4 per WMMA, per half
#define WAVES_PER_BLOCK 4
#define BLOCK_THREADS (WAVES_PER_BLOCK * 32)
#define LDSW_ELEMS (32 * NT * 32 * 2)   // 8192 floats = 32 KB

// ---------------------------------------------------------------------------
// Main kernel: one wave32 per 16-edge tile.
//   D[16x16] = A[16x4] x B[4x16] accumulated over K=128 (32 k-steps), 4 n-tiles.
//   A built from global gathers of x (L2-resident), B from LDS-staged W^T.
// ---------------------------------------------------------------------------
__launch_bounds__(BLOCK_THREADS)
__global__ void edgeconv_wmma(const float* __restrict__ x,
                              const long long* __restrict__ rowIdx,
                              const long long* __restrict__ colIdx,
                              const float* __restrict__ W,
                              const float* __restrict__ bn_gamma,
                              const float* __restrict__ bn_beta,
                              const float* __restrict__ bn_mean,
                              const float* __restrict__ bn_var,
                              float* __restrict__ out,
                              int nTiles)
{
    // W^T fragments laid out exactly as the 32-bit 4x16 B-matrix wants them:
    // element (kk, nt, lane, v) = W[n][k] with n = nt*16 + (lane&15),
    // k = kk*4 + 2*(lane>>4) + v   (lanes 0-15: K=v .. ; lanes 16-31: K=v+2 ..)
    __shared__ float ldsW[LDSW_ELEMS];

    const int tid = threadIdx.x;
    for (int i = tid; i < LDSW_ELEMS; i += BLOCK_THREADS) {
        const int v    = i & 1;
        const int lane = (i >> 1) & 31;
        const int nt   = (i >> 6) & 3;
        const int kk   = i >> 8;                 // 0..31
        const int n    = nt * 16 + (lane & 15);
        const int k    = kk * 4 + ((lane >> 4) << 1) + v;
        ldsW[i] = W[n * TWO_C + k];
    }
    __syncthreads();

    const int lane    = tid & 31;
    const int wave    = tid >> 5;
    const int halfSel = lane >> 4;               // 0 | 1
    const int nlocal  = lane & 15;
    const int koff    = halfSel << 1;            // A-matrix lane K offset

    // BN constants: channel depends only on (lane, nt) -> hoist out of loop
    float bnInv[NT], bnBias[NT];
#pragma unroll
    for (int nt = 0; nt < NT; ++nt) {
        const int n  = nt * 16 + nlocal;
        const float iv = bn_gamma[n] * rsqrtf(bn_var[n] + 1e-5f);
        bnInv[nt]  = iv;
        bnBias[nt] = bn_beta[n] - bn_mean[n] * iv;
    }

    const int totalWaves = gridDim.x * WAVES_PER_BLOCK;
    for (int tile = blockIdx.x * WAVES_PER_BLOCK + wave; tile < nTiles;
         tile += totalWaves) {
        const long long base = (long long)tile * TILE;
        const long long r = rowIdx[base + nlocal];
        const long long c = colIdx[base + nlocal];
        const float* __restrict__ xr = x + r * C_IN;
        const float* __restrict__ xc = x + c * C_IN;

        v8f acc[NT] = {};

        // ---- first half of K: E[:,0:64] = x_i ----
#pragma unroll 4
        for (int kk = 0; kk < KSTEPS; ++kk) {
            v2f a = *(const v2f*)(xr + kk * 4 + koff);
#pragma unroll
            for (int nt = 0; nt < NT; ++nt) {
                v2f b = *(const v2f*)&ldsW[((kk * NT + nt) * 32 + lane) * 2];
                acc[nt] = __builtin_amdgcn_wmma_f32_16x16x4_f32(
                    false, a, false, b, (short)0, acc[nt], false, false);
            }
        }
        // ---- second half of K: E[:,64:128] = x_j - x_i ----
#pragma unroll 4
        for (int kk = 0; kk < KSTEPS; ++kk) {
            v2f ai = *(const v2f*)(xr + kk * 4 + koff);
            v2f aj = *(const v2f*)(xc + kk * 4 + koff);
            v2f a  = aj - ai;
#pragma unroll
            for (int nt = 0; nt < NT; ++nt) {
                v2f b = *(const v2f*)
                    &ldsW[(((kk + KSTEPS) * NT + nt) * 32 + lane) * 2];
                acc[nt] = __builtin_amdgcn_wmma_f32_16x16x4_f32(
                    false, a, false, b, (short)0, acc[nt], false, false);
            }
        }

        // ---- BN + SiLU + scatter-add ----
        // D layout: element v of lane L -> edge m = v + 8*(L>>4),
        //           channel n = nt*16 + (L&15)
#pragma unroll
        for (int v = 0; v < 8; ++v) {
            const int m = v + (halfSel << 3);
            const long long rm = rowIdx[base + m];
            float* __restrict__ op = out + rm * C_OUT + nlocal;
#pragma unroll
            for (int nt = 0; nt < NT; ++nt) {
                float h = acc[nt][v] * bnInv[nt] + bnBias[nt];
                h = h / (1.0f + __expf(-h));        // SiLU
                atomicAdd(op + nt * 16, h);
            }
        }
    }
}

// ---------------------------------------------------------------------------
// Scalar tail kernel for E % 16 edges (kept out of the WMMA kernel so EXEC
// stays all-ones there). One thread per (edge, channel).
// ---------------------------------------------------------------------------
__global__ void edgeconv_tail(const float* __restrict__ x,
                              const long long* __restrict__ rowIdx,
                              const long long* __restrict__ colIdx,
                              const float* __restrict__ W,
                              const float* __restrict__ bn_gamma,
                              const float* __restrict__ bn_beta,
                              const float* __restrict__ bn_mean,
                              const float* __restrict__ bn_var,
                              float* __restrict__ out,
                              long long start, int tailEdges)
{
    const int t = blockIdx.x * blockDim.x + threadIdx.x;
    if (t >= tailEdges * C_OUT) return;
    const int eidx = t >> 6;
    const int n    = t & 63;
    const long long e = start + eidx;
    const long long r = rowIdx[e];
    const long long c = colIdx[e];
    const float* xi = x + r * C_IN;
    const float* xj = x + c * C_IN;
    float s = 0.0f;
    for (int k = 0; k < C_IN; ++k)
        s += W[n * TWO_C + k] * xi[k] + W[n * TWO_C + C_IN + k] * (xj[k] - xi[k]);
    const float iv = bn_gamma[n] * rsqrtf(bn_var[n] + 1e-5f);
    float h = s * iv + (bn_beta[n] - bn_mean[n] * iv);
    h = h / (1.0f + __expf(-h));
    atomicAdd(out + r * C_OUT + n, h);
}

extern "C" void kernel_launch(void* const* d_in, const int* in_sizes, int n_in,
                              void* d_out, int out_size, void* d_ws, size_t ws_size,
                              hipStream_t stream) {
    (void)n_in; (void)d_ws; (void)ws_size;
    const float*     x        = (const float*)d_in[0];
    const long long* ei       = (const long long*)d_in[1];   // int64 [2, E]
    const float*     W        = (const float*)d_in[2];
    const float*     bn_gamma = (const float*)d_in[3];
    const float*     bn_beta  = (const float*)d_in[4];
    const float*     bn_mean  = (const float*)d_in[5];
    const float*     bn_var   = (const float*)d_in[6];
    float*           out      = (float*)d_out;

    const long long E = (long long)in_sizes[1] / 2;
    const long long* rowIdx = ei;
    const long long* colIdx = ei + E;

    // d_out is poisoned by the harness -> zero it (memset node is graph-capturable)
    hipMemsetAsync(d_out, 0, (size_t)out_size * sizeof(float), stream);

    const int nTiles = (int)(E / TILE);
    if (nTiles > 0) {
        int blocks = (nTiles + WAVES_PER_BLOCK - 1) / WAVES_PER_BLOCK;
        if (blocks > 4096) blocks = 4096;
        edgeconv_wmma<<<blocks, BLOCK_THREADS, 0, stream>>>(
            x, rowIdx, colIdx, W, bn_gamma, bn_beta, bn_mean, bn_var, out, nTiles);
    }

    const int tail = (int)(E % TILE);
    if (tail > 0) {
        const int threads = tail * C_OUT;        // <= 1024
        edgeconv_tail<<<(threads + 255) / 256, 256, 0, stream>>>(
            x, rowIdx, colIdx, W, bn_gamma, bn_beta, bn_mean, bn_var, out,
            (long long)nTiles * TILE, tail);
    }
}